// FashionContrastiveLoss_309237645593
// MI455X (gfx1250) — compile-verified
//
#include <hip/hip_runtime.h>

typedef __attribute__((ext_vector_type(2))) float v2f;
typedef __attribute__((ext_vector_type(8))) float v8f;
typedef __attribute__((ext_vector_type(4))) int   v4i;

#define GLOBAL_AS __attribute__((address_space(1)))
#define LDS_AS    __attribute__((address_space(3)))
typedef GLOBAL_AS v4i* gv4i_p;
typedef LDS_AS    v4i* lv4i_p;

#define B_DIM 8192
#define D_DIM 512
#define INV_T 14.285714285714286f   /* 1 / 0.07 */

constexpr int BM = 128, BN = 64, KC = 32, LS = 36;  // LS mult of 4 -> 16B-aligned rows
constexpr int NPR = B_DIM / BN;   // 128 row-partial slices
constexpr int NPC = B_DIM / BM;   // 64  col-partial slices

// ---------------------------------------------------------------------------
// Kernel 1: tiled f32 WMMA GEMM (async-to-LDS double-buffered),
//           fused exp() and per-tile row/col sum partials
// ---------------------------------------------------------------------------
__global__ __launch_bounds__(256) void clip_gemm_exp(const float* __restrict__ img,
                                                     const float* __restrict__ txt,
                                                     float* __restrict__ rowpart,
                                                     float* __restrict__ colpart)
{
    __shared__ float sImg[2][BM][LS];
    __shared__ float sTxt[2][BN][LS];
    __shared__ float sRowP[2][BM];
    __shared__ float sColP[4][BN];

    const int bm   = blockIdx.x, bn = blockIdx.y;
    const int tid  = threadIdx.x;
    const int lane = tid & 31;
    const int wid  = tid >> 5;
    const int wm   = wid & 3;          // 4 row groups of 32 rows
    const int wn   = wid >> 2;         // 2 col groups of 32 cols
    const int lm   = lane & 15;        // M / N index within fragment
    const int kh   = (lane >> 4) << 1; // lanes 16-31 hold K+2,K+3

    const float* gImg = img + (size_t)(bm * BM) * D_DIM;
    const float* gTxt = txt + (size_t)(bn * BN) * D_DIM;

    // per-thread async-copy slices: img 128x32 floats = 1024 xB128 (4/thread),
    //                               txt  64x32 floats =  512 xB128 (2/thread)
    auto load_chunk = [&](int buf, int kt) {
        #pragma unroll
        for (int i = 0; i < 4; ++i) {
            int f = tid + i * 256;
            int r = f >> 3;
            int c = (f & 7) << 2;
            __builtin_amdgcn_global_load_async_to_lds_b128(
                (gv4i_p)(gImg + (size_t)r * D_DIM + kt + c),
                (lv4i_p)&sImg[buf][r][c], 0, 0);
        }
        #pragma unroll
        for (int i = 0; i < 2; ++i) {
            int f = tid + i * 256;
            int r = f >> 3;
            int c = (f & 7) << 2;
            __builtin_amdgcn_global_load_async_to_lds_b128(
                (gv4i_p)(gTxt + (size_t)r * D_DIM + kt + c),
                (lv4i_p)&sTxt[buf][r][c], 0, 0);
        }
    };

    v8f acc00 = {}, acc01 = {}, acc10 = {}, acc11 = {};

    load_chunk(0, 0);
    int buf = 0;
    for (int kt = 0; kt < D_DIM; kt += KC) {
        __builtin_amdgcn_s_wait_asynccnt(0);  // own async writes to LDS done
        __syncthreads();                      // everyone's writes visible
        if (kt + KC < D_DIM) load_chunk(buf ^ 1, kt + KC);
        #pragma unroll
        for (int kc = 0; kc < KC; kc += 4) {
            v2f a0 = *(const v2f*)&sImg[buf][wm * 32 + lm][kc + kh];
            v2f a1 = *(const v2f*)&sImg[buf][wm * 32 + 16 + lm][kc + kh];
            v2f b0 = *(const v2f*)&sTxt[buf][wn * 32 + lm][kc + kh];
            v2f b1 = *(const v2f*)&sTxt[buf][wn * 32 + 16 + lm][kc + kh];
            acc00 = __builtin_amdgcn_wmma_f32_16x16x4_f32(false, a0, false, b0,
                                                          (short)0, acc00, false, false);
            acc01 = __builtin_amdgcn_wmma_f32_16x16x4_f32(false, a0, false, b1,
                                                          (short)0, acc01, false, false);
            acc10 = __builtin_amdgcn_wmma_f32_16x16x4_f32(false, a1, false, b0,
                                                          (short)0, acc10, false, false);
            acc11 = __builtin_amdgcn_wmma_f32_16x16x4_f32(false, a1, false, b1,
                                                          (short)0, acc11, false, false);
        }
        buf ^= 1;
    }

    // ---- epilogue: e = exp(logit/T); row & col partial sums ----
    // C layout: VGPR v, lanes 0-15 -> M=v, N=lane; lanes 16-31 -> M=v+8, N=lane-16
    float e00[8], e01[8], e10[8], e11[8];
    float cs0 = 0.f, cs1 = 0.f;
    #pragma unroll
    for (int v = 0; v < 8; ++v) {
        e00[v] = __expf(acc00[v] * INV_T);
        e01[v] = __expf(acc01[v] * INV_T);
        e10[v] = __expf(acc10[v] * INV_T);
        e11[v] = __expf(acc11[v] * INV_T);
        cs0 += e00[v] + e10[v];   // col frag 0, both row frags
        cs1 += e01[v] + e11[v];   // col frag 1
    }
    cs0 += __shfl_xor(cs0, 16);
    cs1 += __shfl_xor(cs1, 16);
    if (lane < 16) {
        sColP[wm][wn * 32 + lane]      = cs0;
        sColP[wm][wn * 32 + 16 + lane] = cs1;
    }
    #pragma unroll
    for (int v = 0; v < 8; ++v) {
        float r0 = e00[v] + e01[v];   // row frag 0: this wave's 32 columns
        float r1 = e10[v] + e11[v];   // row frag 1
        r0 += __shfl_xor(r0, 1); r0 += __shfl_xor(r0, 2);
        r0 += __shfl_xor(r0, 4); r0 += __shfl_xor(r0, 8);
        r1 += __shfl_xor(r1, 1); r1 += __shfl_xor(r1, 2);
        r1 += __shfl_xor(r1, 4); r1 += __shfl_xor(r1, 8);
        if (lm == 0) {
            int half = (lane >> 4) << 3;
            sRowP[wn][wm * 32 + v + half]      = r0;
            sRowP[wn][wm * 32 + 16 + v + half] = r1;
        }
    }
    __syncthreads();

    if (tid < BM) {
        float rs = sRowP[0][tid] + sRowP[1][tid];
        rowpart[(size_t)bn * B_DIM + (size_t)bm * BM + tid] = rs;
    }
    if (tid < BN) {
        float cs = sColP[0][tid] + sColP[1][tid] + sColP[2][tid] + sColP[3][tid];
        colpart[(size_t)bm * B_DIM + (size_t)bn * BN + tid] = cs;
    }
}

// ---------------------------------------------------------------------------
// Kernel 2: fixed-order reduction of the partials per row/col, then log()
// ---------------------------------------------------------------------------
__global__ __launch_bounds__(256) void clip_reduce_lse(const float* __restrict__ rowpart,
                                                       const float* __restrict__ colpart,
                                                       float* __restrict__ rowlse,
                                                       float* __restrict__ collse)
{
    int i = blockIdx.x * blockDim.x + threadIdx.x;
    if (i >= B_DIM) return;
    float rs = 0.f;
    for (int p = 0; p < NPR; ++p) rs += rowpart[(size_t)p * B_DIM + i];
    float cs = 0.f;
    for (int p = 0; p < NPC; ++p) cs += colpart[(size_t)p * B_DIM + i];
    rowlse[i] = __logf(rs);
    collse[i] = __logf(cs);
}

// ---------------------------------------------------------------------------
// Kernel 3: diagonal dots + final scalar loss (single block, fixed order)
// ---------------------------------------------------------------------------
__global__ __launch_bounds__(256) void clip_finalize(const float* __restrict__ img,
                                                     const float* __restrict__ txt,
                                                     const float* __restrict__ rowlse,
                                                     const float* __restrict__ collse,
                                                     float* __restrict__ out)
{
    __shared__ float red[256];
    float acc = 0.f;
    for (int i = threadIdx.x; i < B_DIM; i += 256) {
        const float4* pi = (const float4*)(img + (size_t)i * D_DIM);
        const float4* pt = (const float4*)(txt + (size_t)i * D_DIM);
        float d = 0.f;
        #pragma unroll 4
        for (int k = 0; k < D_DIM / 4; ++k) {
            float4 a = pi[k], b = pt[k];
            d += a.x * b.x + a.y * b.y + a.z * b.z + a.w * b.w;
        }
        acc += rowlse[i] + collse[i] - 2.f * d * INV_T;
    }
    red[threadIdx.x] = acc;
    __syncthreads();
    for (int s = 128; s > 0; s >>= 1) {
        if (threadIdx.x < s) red[threadIdx.x] += red[threadIdx.x + s];
        __syncthreads();
    }
    if (threadIdx.x == 0) out[0] = red[0] / (2.0f * B_DIM);
}

// ---------------------------------------------------------------------------
extern "C" void kernel_launch(void* const* d_in, const int* in_sizes, int n_in,
                              void* d_out, int out_size, void* d_ws, size_t ws_size,
                              hipStream_t stream)
{
    (void)in_sizes; (void)n_in; (void)out_size; (void)ws_size;
    const float* img = (const float*)d_in[0];
    const float* txt = (const float*)d_in[1];
    float* ws = (float*)d_ws;

    // ws (floats): rowpart[128*8192] | colpart[64*8192] | rowlse[8192] | collse[8192]
    float* rowpart = ws;
    float* colpart = rowpart + (size_t)NPR * B_DIM;
    float* rowlse  = colpart + (size_t)NPC * B_DIM;
    float* collse  = rowlse + B_DIM;

    dim3 grid(B_DIM / BM, B_DIM / BN);   // 64 x 128 tiles
    clip_gemm_exp<<<grid, 256, 0, stream>>>(img, txt, rowpart, colpart);
    clip_reduce_lse<<<(B_DIM + 255) / 256, 256, 0, stream>>>(rowpart, colpart, rowlse, collse);
    clip_finalize<<<1, 256, 0, stream>>>(img, txt, rowlse, collse, (float*)d_out);
}